// Elastic_40656160424250
// MI455X (gfx1250) — compile-verified
//
#include <hip/hip_runtime.h>
#include <hip/hip_bf16.h>

// ---------------- problem constants (match reference) ----------------
constexpr int   NYr   = 300, NXr = 300;
constexpr int   PMLW  = 20;
constexpr int   NYP   = NYr + 2 * PMLW;       // 340
constexpr int   NXP   = NXr + 2 * PMLW;       // 340
constexpr int   NT    = 128;
constexpr int   NSHOT = 2, NSRC = 1, NREC = 32;
constexpr float DXC   = 5.0f;
constexpr float DTC   = 0.001f;
constexpr float C1    = 9.0f / 8.0f;
constexpr float C2    = -1.0f / 24.0f;
constexpr float VMAXC = 2000.0f;
constexpr float RPML  = 1e-6f;
constexpr float INVDX = 1.0f / DXC;

constexpr int NPP = NYP * NXP;                // 115600 cells per plane

// ---------------- workspace layout (floats) ----------------
constexpr int WS_LAM  = 0;
constexpr int WS_MU   = NPP;
constexpr int WS_BUO  = 2 * NPP;
constexpr int WS_LP2M = 3 * NPP;
constexpr int WS_BY   = 4 * NPP;              // NYP entries
constexpr int WS_BX   = 4 * NPP + 512;        // NXP entries
constexpr int WS_F    = 4 * NPP + 1024;       // 26 field planes follow

// field ids
enum {
  F_VY = 0, F_VX, F_SYY, F_SXX, F_SXY,
  F_MVYY, F_MVYX, F_MVXY, F_MVXX,
  F_MSYYY, F_MSXYX, F_MSXYY, F_MSXXX,
  F_COUNT
};

__device__ __forceinline__ float* fld(float* ws, int f, int s) {
  return ws + WS_F + (size_t)(f * NSHOT + s) * NPP;
}
__device__ __forceinline__ const float* fldc(const float* ws, int f, int s) {
  return ws + WS_F + (size_t)(f * NSHOT + s) * NPP;
}

// ---------------- CDNA5 async global->LDS path (probe-verified) ----------------
#if __has_builtin(__builtin_amdgcn_global_load_async_to_lds_b32) && \
    __has_builtin(__builtin_amdgcn_s_wait_asynccnt)
#define HAVE_ASYNC_LDS 1
typedef __attribute__((address_space(1))) int* GlobalIntPtr;
typedef __attribute__((address_space(3))) int* LdsIntPtr;
#endif

__device__ __forceinline__ void async_copy_f32(float* ldsDst, const float* gSrc) {
#ifdef HAVE_ASYNC_LDS
  // ASYNCcnt-tracked DMA-style copy L2 -> LDS, no VGPR round-trip.
  __builtin_amdgcn_global_load_async_to_lds_b32(
      (GlobalIntPtr)(void*)gSrc, (LdsIntPtr)(void*)ldsDst, 0, 0);
#else
  *ldsDst = *gSrc;
#endif
}

__device__ __forceinline__ void async_fence_barrier() {
#ifdef HAVE_ASYNC_LDS
  __builtin_amdgcn_s_wait_asynccnt(0);   // this wave's async fills complete
#endif
  __syncthreads();                        // all waves' fills visible
}

// ---------------- temporal blocking geometry ----------------
constexpr int TSTEPS = 4;                 // time steps fused per launch
constexpr int TBO    = 32;                // output tile edge
constexpr int MARG   = 4 * TSTEPS;        // 16: halo per side (2 stages x 2 cells x TSTEPS)
constexpr int EXT    = TBO + 2 * MARG;    // 64: extended tile edge (power of two)
constexpr int EXT_SH = 6;                 // log2(EXT)
constexpr int EXT_MK = EXT - 1;
constexpr int NTHR   = 512;               // 16 wave32 waves
constexpr int CPT    = (EXT * EXT) / NTHR;// 8 cells per thread (4096/512)

// ---------------- init: pad model, decay profiles, zero state ----------------
__global__ __launch_bounds__(256) void ElasticInit(
    float* __restrict__ ws, const float* __restrict__ lamb,
    const float* __restrict__ mu, const float* __restrict__ buo) {
  int gid = blockIdx.x * blockDim.x + threadIdx.x;
  if (gid < NPP) {
    int y = gid / NXP, x = gid - y * NXP;
    int iy = min(max(y - PMLW, 0), NYr - 1);
    int ix = min(max(x - PMLW, 0), NXr - 1);
    float la = lamb[iy * NXr + ix];
    float m  = mu[iy * NXr + ix];
    ws[WS_LAM + gid]  = la;
    ws[WS_MU + gid]   = m;
    ws[WS_BUO + gid]  = buo[iy * NXr + ix];
    ws[WS_LP2M + gid] = la + 2.0f * m;
    for (int f = 0; f < F_COUNT * NSHOT; ++f)
      ws[WS_F + (size_t)f * NPP + gid] = 0.0f;
  }
  if (gid < NYP) {  // NYP == NXP: one decay profile serves both axes
    float d0 = 3.0f * VMAXC / (2.0f * PMLW * DXC) * logf(1.0f / RPML);
    float fi = (float)gid;
    float lo = fminf(fmaxf(((float)PMLW - fi) / (float)PMLW, 0.0f), 1.0f);
    float hi = fminf(fmaxf((fi - (float)(NYP - 1 - PMLW)) / (float)PMLW, 0.0f), 1.0f);
    float mm = fmaxf(lo, hi);
    float b  = expf(-d0 * mm * mm * DTC);
    ws[WS_BY + gid] = b;
    ws[WS_BX + gid] = b;
  }
}

// ---------------- fused 4-step velocity/stress kernel ----------------
__global__ __launch_bounds__(NTHR) void ElasticFused(
    float* __restrict__ ws, const float* __restrict__ samp,
    const int* __restrict__ sloc, const int* __restrict__ rloc,
    float* __restrict__ out, int t0) {
  const int s   = blockIdx.z;
  const int Y0  = blockIdx.y * TBO, X0 = blockIdx.x * TBO;
  const int EY0 = Y0 - MARG,        EX0 = X0 - MARG;
  const int tid = threadIdx.x;

  __shared__ float sVY[EXT][EXT];
  __shared__ float sVX[EXT][EXT];
  __shared__ float sSYY[EXT][EXT];
  __shared__ float sSXX[EXT][EXT];
  __shared__ float sSXY[EXT][EXT];

  // register-resident C-PML memory fields (own-cell access only)
  float m_syyy[CPT], m_sxyx[CPT], m_sxyy[CPT], m_sxxx[CPT];
  float m_vyy[CPT], m_vxx[CPT], m_vyx[CPT], m_vxy[CPT];
  // register-cached PML decay factors per owned cell
  float byk[CPT], bxk[CPT];

  const float* g_vy  = fldc(ws, F_VY, s);
  const float* g_vx  = fldc(ws, F_VX, s);
  const float* g_syy = fldc(ws, F_SYY, s);
  const float* g_sxx = fldc(ws, F_SXX, s);
  const float* g_sxy = fldc(ws, F_SXY, s);

  // -------- stage extended tile: 5 fields async -> LDS, m-fields -> regs ------
  #pragma unroll
  for (int k = 0; k < CPT; ++k) {
    int c  = tid + k * NTHR;
    int ey = c >> EXT_SH, ex = c & EXT_MK;
    int gy = EY0 + ey, gx = EX0 + ex;
    bool in = (gy >= 0) && (gy < NYP) && (gx >= 0) && (gx < NXP);
    if (in) {
      int gi = gy * NXP + gx;
      async_copy_f32(&sVY[ey][ex],  g_vy  + gi);
      async_copy_f32(&sVX[ey][ex],  g_vx  + gi);
      async_copy_f32(&sSYY[ey][ex], g_syy + gi);
      async_copy_f32(&sSXX[ey][ex], g_sxx + gi);
      async_copy_f32(&sSXY[ey][ex], g_sxy + gi);
      m_syyy[k] = fldc(ws, F_MSYYY, s)[gi];
      m_sxyx[k] = fldc(ws, F_MSXYX, s)[gi];
      m_sxyy[k] = fldc(ws, F_MSXYY, s)[gi];
      m_sxxx[k] = fldc(ws, F_MSXXX, s)[gi];
      m_vyy[k]  = fldc(ws, F_MVYY, s)[gi];
      m_vxx[k]  = fldc(ws, F_MVXX, s)[gi];
      m_vyx[k]  = fldc(ws, F_MVYX, s)[gi];
      m_vxy[k]  = fldc(ws, F_MVXY, s)[gi];
      byk[k] = ws[WS_BY + gy];
      bxk[k] = ws[WS_BX + gx];
    } else {
      sVY[ey][ex] = 0.0f; sVX[ey][ex] = 0.0f;
      sSYY[ey][ex] = 0.0f; sSXX[ey][ex] = 0.0f; sSXY[ey][ex] = 0.0f;
      m_syyy[k] = 0.0f; m_sxyx[k] = 0.0f; m_sxyy[k] = 0.0f; m_sxxx[k] = 0.0f;
      m_vyy[k] = 0.0f; m_vxx[k] = 0.0f; m_vyx[k] = 0.0f; m_vxy[k] = 0.0f;
      byk[k] = 1.0f; bxk[k] = 1.0f;
    }
  }
  async_fence_barrier();

  const int sy = sloc[s * 2 + 0] + PMLW;   // NSRC == 1
  const int sx = sloc[s * 2 + 1] + PMLW;

  for (int j = 0; j < TSTEPS; ++j) {
    // ---------------- velocity stage: valid on margin 4j+2 ----------------
    const int mv = 4 * j + 2;
    #pragma unroll
    for (int k = 0; k < CPT; ++k) {
      int c  = tid + k * NTHR;
      int ey = c >> EXT_SH, ex = c & EXT_MK;
      int gy = EY0 + ey, gx = EX0 + ex;
      bool act = (ey >= mv) && (ey < EXT - mv) && (ex >= mv) && (ex < EXT - mv) &&
                 (gy >= 0) && (gy < NYP) && (gx >= 0) && (gx < NXP);
      if (act) {
        const float byv = byk[k];
        const float bxv = bxk[k];
        float d, m;
        // d_plus(syy, y), C-PML msyy_y (by)
        d = (C1 * (sSYY[ey + 1][ex] - sSYY[ey][ex]) +
             C2 * (sSYY[ey + 2][ex] - sSYY[ey - 1][ex])) * INVDX;
        m = byv * m_syyy[k] + (byv - 1.0f) * d; m_syyy[k] = m;
        const float dsyy_dy = d + m;
        // d_minus(sxy, x), msxy_x (bx)
        d = (C1 * (sSXY[ey][ex] - sSXY[ey][ex - 1]) +
             C2 * (sSXY[ey][ex + 1] - sSXY[ey][ex - 2])) * INVDX;
        m = bxv * m_sxyx[k] + (bxv - 1.0f) * d; m_sxyx[k] = m;
        const float dsxy_dx = d + m;
        const float buov = ws[WS_BUO + gy * NXP + gx];
        float v = sVY[ey][ex] + DTC * buov * (dsyy_dy + dsxy_dx);
        if (gy == sy && gx == sx) v += samp[s * NSRC * NT + (t0 + j)];
        sVY[ey][ex] = v;
        // d_minus(sxy, y), msxy_y (by)
        d = (C1 * (sSXY[ey][ex] - sSXY[ey - 1][ex]) +
             C2 * (sSXY[ey + 1][ex] - sSXY[ey - 2][ex])) * INVDX;
        m = byv * m_sxyy[k] + (byv - 1.0f) * d; m_sxyy[k] = m;
        const float dsxy_dy = d + m;
        // d_plus(sxx, x), msxx_x (bx)
        d = (C1 * (sSXX[ey][ex + 1] - sSXX[ey][ex]) +
             C2 * (sSXX[ey][ex + 2] - sSXX[ey][ex - 1])) * INVDX;
        m = bxv * m_sxxx[k] + (bxv - 1.0f) * d; m_sxxx[k] = m;
        const float dsxx_dx = d + m;
        sVX[ey][ex] = sVX[ey][ex] + DTC * buov * (dsxy_dy + dsxx_dx);
      }
    }
    __syncthreads();

    // receiver recording for step t0+j (vy final for this step; unique owner tile)
    if (tid < NREC) {
      int ry = rloc[(s * NREC + tid) * 2 + 0] + PMLW;
      int rx = rloc[(s * NREC + tid) * 2 + 1] + PMLW;
      if (ry >= Y0 && ry < Y0 + TBO && rx >= X0 && rx < X0 + TBO)
        out[(size_t)(s * NREC + tid) * NT + (t0 + j)] = sVY[ry - EY0][rx - EX0];
    }

    // ---------------- stress stage: valid on margin 4j+4 ----------------
    const int ms = 4 * j + 4;
    #pragma unroll
    for (int k = 0; k < CPT; ++k) {
      int c  = tid + k * NTHR;
      int ey = c >> EXT_SH, ex = c & EXT_MK;
      int gy = EY0 + ey, gx = EX0 + ex;
      bool act = (ey >= ms) && (ey < EXT - ms) && (ex >= ms) && (ex < EXT - ms) &&
                 (gy >= 0) && (gy < NYP) && (gx >= 0) && (gx < NXP);
      if (act) {
        const int gi = gy * NXP + gx;
        const float byv = byk[k];
        const float bxv = bxk[k];
        float d, m;
        // d_minus(vy, y), mvyy (by)
        d = (C1 * (sVY[ey][ex] - sVY[ey - 1][ex]) +
             C2 * (sVY[ey + 1][ex] - sVY[ey - 2][ex])) * INVDX;
        m = byv * m_vyy[k] + (byv - 1.0f) * d; m_vyy[k] = m;
        const float dvy_dy = d + m;
        // d_minus(vx, x), mvxx (bx)
        d = (C1 * (sVX[ey][ex] - sVX[ey][ex - 1]) +
             C2 * (sVX[ey][ex + 1] - sVX[ey][ex - 2])) * INVDX;
        m = bxv * m_vxx[k] + (bxv - 1.0f) * d; m_vxx[k] = m;
        const float dvx_dx = d + m;
        const float lp2m = ws[WS_LP2M + gi];
        const float lam  = ws[WS_LAM + gi];
        sSYY[ey][ex] = sSYY[ey][ex] + DTC * (lp2m * dvy_dy + lam * dvx_dx);
        sSXX[ey][ex] = sSXX[ey][ex] + DTC * (lp2m * dvx_dx + lam * dvy_dy);
        // d_plus(vy, x), mvyx (bx)
        d = (C1 * (sVY[ey][ex + 1] - sVY[ey][ex]) +
             C2 * (sVY[ey][ex + 2] - sVY[ey][ex - 1])) * INVDX;
        m = bxv * m_vyx[k] + (bxv - 1.0f) * d; m_vyx[k] = m;
        const float dvy_dx = d + m;
        // d_plus(vx, y), mvxy (by)
        d = (C1 * (sVX[ey + 1][ex] - sVX[ey][ex]) +
             C2 * (sVX[ey + 2][ex] - sVX[ey - 1][ex])) * INVDX;
        m = byv * m_vxy[k] + (byv - 1.0f) * d; m_vxy[k] = m;
        const float dvx_dy = d + m;
        sSXY[ey][ex] = sSXY[ey][ex] + DTC * ws[WS_MU + gi] * (dvy_dx + dvx_dy);
      }
    }
    __syncthreads();
  }

  // -------- write back interior (margin MARG): 5 LDS fields + 8 reg m-fields --
  #pragma unroll
  for (int k = 0; k < CPT; ++k) {
    int c  = tid + k * NTHR;
    int ey = c >> EXT_SH, ex = c & EXT_MK;
    int gy = EY0 + ey, gx = EX0 + ex;
    bool wr = (ey >= MARG) && (ey < EXT - MARG) && (ex >= MARG) && (ex < EXT - MARG) &&
              (gy >= 0) && (gy < NYP) && (gx >= 0) && (gx < NXP);
    if (wr) {
      int gi = gy * NXP + gx;
      fld(ws, F_VY, s)[gi]  = sVY[ey][ex];
      fld(ws, F_VX, s)[gi]  = sVX[ey][ex];
      fld(ws, F_SYY, s)[gi] = sSYY[ey][ex];
      fld(ws, F_SXX, s)[gi] = sSXX[ey][ex];
      fld(ws, F_SXY, s)[gi] = sSXY[ey][ex];
      fld(ws, F_MSYYY, s)[gi] = m_syyy[k];
      fld(ws, F_MSXYX, s)[gi] = m_sxyx[k];
      fld(ws, F_MSXYY, s)[gi] = m_sxyy[k];
      fld(ws, F_MSXXX, s)[gi] = m_sxxx[k];
      fld(ws, F_MVYY, s)[gi]  = m_vyy[k];
      fld(ws, F_MVXX, s)[gi]  = m_vxx[k];
      fld(ws, F_MVYX, s)[gi]  = m_vyx[k];
      fld(ws, F_MVXY, s)[gi]  = m_vxy[k];
    }
  }
}

// ---------------- host driver ----------------
extern "C" void kernel_launch(void* const* d_in, const int* in_sizes, int n_in,
                              void* d_out, int out_size, void* d_ws, size_t ws_size,
                              hipStream_t stream) {
  const float* lamb = (const float*)d_in[0];
  const float* mu   = (const float*)d_in[1];
  const float* buo  = (const float*)d_in[2];
  const float* samp = (const float*)d_in[3];
  const int*   sloc = (const int*)d_in[4];
  const int*   rloc = (const int*)d_in[5];
  float* ws  = (float*)d_ws;
  float* out = (float*)d_out;

  ElasticInit<<<(NPP + 255) / 256, 256, 0, stream>>>(ws, lamb, mu, buo);

  dim3 blk(NTHR, 1, 1);
  dim3 grd((NXP + TBO - 1) / TBO, (NYP + TBO - 1) / TBO, NSHOT);  // 11 x 11 x 2
  for (int t0 = 0; t0 < NT; t0 += TSTEPS) {
    ElasticFused<<<grd, blk, 0, stream>>>(ws, samp, sloc, rloc, out, t0);
  }
}